// YOLOInference_78443282695039
// MI455X (gfx1250) — compile-verified
//
#include <hip/hip_runtime.h>
#include <hip/hip_bf16.h>

#ifndef __has_builtin
#define __has_builtin(x) 0
#endif

#define NUM_B     32
#define NL        17328      // 76*76*3
#define NM        4332       // 38*38*3
#define NS        1083       // 19*19*3
#define NTOT      22743      // NL+NM+NS
#define NPAD      32768
#define TOPK_N    2000
#define MAX_DET   100

#if __has_builtin(__builtin_amdgcn_global_load_async_to_lds_b64) && \
    __has_builtin(__builtin_amdgcn_s_wait_asynccnt)
#define HAVE_ASYNC_LDS 1
#else
#define HAVE_ASYNC_LDS 0
#endif

typedef int v2i __attribute__((ext_vector_type(2)));
typedef __attribute__((address_space(1))) v2i* as1_v2i_ptr;
typedef __attribute__((address_space(3))) v2i* as3_v2i_ptr;

__device__ __forceinline__ float sigf(float x) {
    return 1.0f / (1.0f + __expf(-x));
}

// ---------------------------------------------------------------------------
// Kernel 1: decode all boxes, pack (score,idx) keys, pad tail with zero keys.
// Memory-bound: streams 247 MB of features (~11 us floor at 23.3 TB/s).
// ---------------------------------------------------------------------------
__global__ void yolo_decode_kernel(const float* __restrict__ fl,
                                   const float* __restrict__ fm,
                                   const float* __restrict__ fs,
                                   const float* __restrict__ anchors,
                                   float4* __restrict__ boxes,
                                   int* __restrict__ clsArr,
                                   unsigned long long* __restrict__ keys) {
    int t = blockIdx.x * blockDim.x + threadIdx.x;
    int b = t / NPAD;
    int i = t - b * NPAD;
    if (b >= NUM_B) return;
    if (i >= NTOT) {                       // padding entries: smallest key
        keys[(size_t)b * NPAD + i] = 0ull;
        return;
    }
    const float* f;
    int Wd, off, abase;
    if (i < NL)            { f = fl; Wd = 76; off = 0;       abase = 0; }
    else if (i < NL + NM)  { f = fm; Wd = 38; off = NL;      abase = 3; }
    else                   { f = fs; Wd = 19; off = NL + NM; abase = 6; }

    int li   = i - off;
    int cell = li / 3;
    int a    = li - cell * 3;
    int y    = cell / Wd;
    int x    = cell - y * Wd;

    const float* p = f + (((size_t)b * Wd + y) * Wd + x) * 255 + a * 85;
    __builtin_prefetch(p, 0, 0);           // -> global_prefetch_b8 (gfx1250)

    float tx = p[0], ty = p[1], tw = p[2], th = p[3], tc = p[4];

    // class-max over logits (sigmoid monotone => max prob == sigmoid(max logit))
    float ml = p[5];
    int   mc = 0;
#pragma unroll 8
    for (int c = 1; c < 80; ++c) {
        float v = p[5 + c];
        if (v > ml) { ml = v; mc = c; }
    }
    float score = sigf(tc) * sigf(ml);

    float aw = anchors[(abase + a) * 2 + 0];
    float ah = anchors[(abase + a) * 2 + 1];
    float invW = 1.0f / (float)Wd;
    float cx = (sigf(tx) + (float)x) * invW;
    float cy = (sigf(ty) + (float)y) * invW;
    float hw = 0.5f * __expf(tw) * aw * (1.0f / 608.0f);
    float hh = 0.5f * __expf(th) * ah * (1.0f / 608.0f);

    float4 bx;
    bx.x = cx - hw; bx.y = cy - hh; bx.z = cx + hw; bx.w = cy + hh;

    size_t bi = (size_t)b * NTOT + i;
    boxes[bi]  = bx;
    clsArr[bi] = mc;

    unsigned sb = __float_as_uint(score);  // score > 0 => bit pattern order-preserving
    keys[(size_t)b * NPAD + i] = ((unsigned long long)sb << 32) | (unsigned)i;
}

// ---------------------------------------------------------------------------
// Kernel 2: per-image bitonic sort (descending) of 32768 u64 keys,
// performed ENTIRELY IN LDS (256 KB of the WGP's 320 KB), staged in/out with
// the CDNA5 async global<->LDS path (ASYNCcnt).
// One 1024-thread workgroup per image; 32 images -> 32 WGPs.
// ---------------------------------------------------------------------------
__global__ void __launch_bounds__(1024)
yolo_sort_kernel(unsigned long long* __restrict__ keys) {
    __shared__ unsigned long long sk[NPAD];        // 262144 B
    unsigned long long* g = keys + (size_t)blockIdx.x * NPAD;

#if HAVE_ASYNC_LDS
    for (unsigned i = threadIdx.x; i < NPAD; i += blockDim.x) {
        __builtin_amdgcn_global_load_async_to_lds_b64(
            (as1_v2i_ptr)(void*)(g + i),
            (as3_v2i_ptr)(void*)&sk[i],
            0, 0);
    }
    __builtin_amdgcn_s_wait_asynccnt(0);
#else
    for (unsigned i = threadIdx.x; i < NPAD; i += blockDim.x) sk[i] = g[i];
#endif
    __syncthreads();

    for (unsigned k = 2; k <= NPAD; k <<= 1) {
        for (unsigned j = k >> 1; j > 0; j >>= 1) {
            for (unsigned i = threadIdx.x; i < NPAD; i += blockDim.x) {
                unsigned ixj = i ^ j;
                if (ixj > i) {
                    unsigned long long va = sk[i];
                    unsigned long long vb = sk[ixj];
                    bool desc = ((i & k) == 0);
                    if (desc ? (va < vb) : (va > vb)) {
                        sk[i]   = vb;
                        sk[ixj] = va;
                    }
                }
            }
            __syncthreads();
        }
    }

    // Stage back out (NMS consumes the head; write the whole array so the
    // workspace state is deterministic).
#if HAVE_ASYNC_LDS && __has_builtin(__builtin_amdgcn_global_store_async_from_lds_b64)
    for (unsigned i = threadIdx.x; i < NPAD; i += blockDim.x) {
        __builtin_amdgcn_global_store_async_from_lds_b64(
            (as1_v2i_ptr)(void*)(g + i),
            (as3_v2i_ptr)(void*)&sk[i],
            0, 0);
    }
    __builtin_amdgcn_s_wait_asynccnt(0);
#else
    for (unsigned i = threadIdx.x; i < NPAD; i += blockDim.x) g[i] = sk[i];
#endif
}

// ---------------------------------------------------------------------------
// Kernel 3: per-image greedy NMS over the sorted top-2000.
// LDS ~64 KB/WG (well under the 320 KB WGP budget).
// ---------------------------------------------------------------------------
__global__ void yolo_nms_kernel(const unsigned long long* __restrict__ keys,
                                const float4* __restrict__ boxes,
                                const int* __restrict__ clsArr,
                                float* __restrict__ out) {
    __shared__ unsigned long long skeys[TOPK_N];   // 16000 B
    __shared__ float4 snb[TOPK_N];                 // 32000 B (class-offset boxes)
    __shared__ float  sarea[TOPK_N];               //  8000 B
    __shared__ float  sscore[TOPK_N];              //  8000 B
    __shared__ int    scur, ssel;

    const int b = blockIdx.x;
    const unsigned long long* gk = keys + (size_t)b * NPAD;

    // Stage sorted keys into LDS via the CDNA5 async global->LDS path.
#if HAVE_ASYNC_LDS
    for (int i = threadIdx.x; i < TOPK_N; i += blockDim.x) {
        __builtin_amdgcn_global_load_async_to_lds_b64(
            (as1_v2i_ptr)(void*)(gk + i),
            (as3_v2i_ptr)(void*)&skeys[i],
            0, 0);
    }
    __builtin_amdgcn_s_wait_asynccnt(0);
#else
    for (int i = threadIdx.x; i < TOPK_N; i += blockDim.x) skeys[i] = gk[i];
#endif
    __syncthreads();

    // Gather class-offset boxes, areas, scores.
    for (int i = threadIdx.x; i < TOPK_N; i += blockDim.x) {
        unsigned long long kv = skeys[i];
        unsigned idx = (unsigned)kv;
        float sc = __uint_as_float((unsigned)(kv >> 32));
        float4 bx = boxes[(size_t)b * NTOT + idx];
        float co = 2.0f * (float)clsArr[(size_t)b * NTOT + idx];
        bx.x += co; bx.y += co; bx.z += co; bx.w += co;
        snb[i]    = bx;
        sarea[i]  = (bx.z - bx.x) * (bx.w - bx.y);
        sscore[i] = sc;
    }
    if (threadIdx.x == 0) scur = 0;
    __syncthreads();

    float* orow = out + (size_t)b * MAX_DET * 6;

    for (int d = 0; d < MAX_DET; ++d) {
        if (threadIdx.x == 0) {
            int c = scur;
            while (c < TOPK_N && sscore[c] < 0.0f) ++c;
            scur = c;
            ssel = (c < TOPK_N) ? c : -1;
            float* r = orow + d * 6;
            if (ssel >= 0) {
                unsigned long long kv = skeys[c];
                unsigned idx = (unsigned)kv;
                float4 rb = boxes[(size_t)b * NTOT + idx];
                r[0] = rb.x; r[1] = rb.y; r[2] = rb.z; r[3] = rb.w;
                r[4] = __uint_as_float((unsigned)(kv >> 32));
                r[5] = (float)clsArr[(size_t)b * NTOT + idx];
            } else {
                r[0] = 0.f; r[1] = 0.f; r[2] = 0.f;
                r[3] = 0.f; r[4] = 0.f; r[5] = 0.f;
            }
        }
        __syncthreads();
        int sel = ssel;
        if (sel >= 0) {
            float4 sb4 = snb[sel];
            float sa = sarea[sel];
            for (int i = threadIdx.x; i < TOPK_N; i += blockDim.x) {
                float4 o = snb[i];
                float xx1 = fmaxf(sb4.x, o.x);
                float yy1 = fmaxf(sb4.y, o.y);
                float xx2 = fminf(sb4.z, o.z);
                float yy2 = fminf(sb4.w, o.w);
                float inter = fmaxf(xx2 - xx1, 0.0f) * fmaxf(yy2 - yy1, 0.0f);
                float iou = inter / (sa + sarea[i] - inter + 1e-9f);
                if (iou > 0.5f) sscore[i] = -1.0f;   // suppress (incl. self)
            }
        }
        __syncthreads();
    }
}

// ---------------------------------------------------------------------------
extern "C" void kernel_launch(void* const* d_in, const int* in_sizes, int n_in,
                              void* d_out, int out_size, void* d_ws, size_t ws_size,
                              hipStream_t stream) {
    const float* fl      = (const float*)d_in[0];
    const float* fm      = (const float*)d_in[1];
    const float* fs      = (const float*)d_in[2];
    const float* anchors = (const float*)d_in[3];
    float* out = (float*)d_out;

    // Workspace layout (all 16B-aligned):
    //   keys : 32 * 32768 * 8  = 8,388,608 B
    //   boxes: 32 * 22743 * 16 = 11,644,416 B
    //   cls  : 32 * 22743 * 4  = 2,911,104 B            (total ~22.9 MB)
    char* ws = (char*)d_ws;
    unsigned long long* keys = (unsigned long long*)ws;
    float4* boxes = (float4*)(ws + (size_t)NUM_B * NPAD * 8);
    int* cls = (int*)(ws + (size_t)NUM_B * NPAD * 8 + (size_t)NUM_B * NTOT * 16);

    dim3 dblk(256);
    dim3 dgrid((NUM_B * NPAD) / 256);
    yolo_decode_kernel<<<dgrid, dblk, 0, stream>>>(fl, fm, fs, anchors,
                                                   boxes, cls, keys);

    yolo_sort_kernel<<<NUM_B, 1024, 0, stream>>>(keys);

    yolo_nms_kernel<<<NUM_B, 256, 0, stream>>>(keys, boxes, cls, out);
}